// MC_Module_Batch_86938728005894
// MI455X (gfx1250) — compile-verified
//
#include <hip/hip_runtime.h>
#include <stdint.h>

// Problem constants (from the reference)
#define Bb 4
#define Rr 2
#define Cc 19
#define Hh 256
#define Ww 512
#define HW (Hh * Ww)

// One thread per output pixel (b, h, w). Bilinear taps + weights are computed
// once per (pixel, ref) and reused across all 19 channels. mv/weight are
// staged global->LDS with CDNA5 async copies (ASYNCcnt), pred is gathered
// through L1/L2 (whole working set fits in the 192MB L2).
__global__ __launch_bounds__(256, 8)
void mc_warp_sum_kernel(const float* __restrict__ pred,
                        const float* __restrict__ mv,
                        const float* __restrict__ wgt,
                        float* __restrict__ out) {
    const int tid = threadIdx.x;
    const int w   = blockIdx.x * 256 + tid;
    const int h   = blockIdx.y;
    const int b   = blockIdx.z;
    const uint32_t pix = (uint32_t)(h * Ww + w);

    // ---------- Stage mv(x,y) and weight for both refs into LDS ----------
    // CDNA5 async global->LDS copy: each lane DMAs its own dword; consecutive
    // lanes -> coalesced 128B global requests, conflict-free LDS banks.
    __shared__ float smem[6 * 256];
    {
        const uint64_t mv_base = (uint64_t)(uintptr_t)mv;
        const uint64_t wt_base = (uint64_t)(uintptr_t)wgt;

        // LDS byte addresses (low 32 bits of the generic pointer = LDS offset)
        uint32_t l_mx0 = (uint32_t)(uintptr_t)(&smem[0 * 256 + tid]);
        uint32_t l_my0 = (uint32_t)(uintptr_t)(&smem[1 * 256 + tid]);
        uint32_t l_mx1 = (uint32_t)(uintptr_t)(&smem[2 * 256 + tid]);
        uint32_t l_my1 = (uint32_t)(uintptr_t)(&smem[3 * 256 + tid]);
        uint32_t l_wt0 = (uint32_t)(uintptr_t)(&smem[4 * 256 + tid]);
        uint32_t l_wt1 = (uint32_t)(uintptr_t)(&smem[5 * 256 + tid]);

        // Global byte offsets
        uint32_t g_mx0 = (uint32_t)((((b * Rr + 0) * 2 + 0) * HW + pix)) * 4u;
        uint32_t g_my0 = (uint32_t)((((b * Rr + 0) * 2 + 1) * HW + pix)) * 4u;
        uint32_t g_mx1 = (uint32_t)((((b * Rr + 1) * 2 + 0) * HW + pix)) * 4u;
        uint32_t g_my1 = (uint32_t)((((b * Rr + 1) * 2 + 1) * HW + pix)) * 4u;
        uint32_t g_wt0 = (uint32_t)(((b * Rr + 0) * HW + pix)) * 4u;
        uint32_t g_wt1 = (uint32_t)(((b * Rr + 1) * HW + pix)) * 4u;

        asm volatile("global_load_async_to_lds_b32 %0, %1, %2"
                     :: "v"(l_mx0), "v"(g_mx0), "s"(mv_base) : "memory");
        asm volatile("global_load_async_to_lds_b32 %0, %1, %2"
                     :: "v"(l_my0), "v"(g_my0), "s"(mv_base) : "memory");
        asm volatile("global_load_async_to_lds_b32 %0, %1, %2"
                     :: "v"(l_mx1), "v"(g_mx1), "s"(mv_base) : "memory");
        asm volatile("global_load_async_to_lds_b32 %0, %1, %2"
                     :: "v"(l_my1), "v"(g_my1), "s"(mv_base) : "memory");
        asm volatile("global_load_async_to_lds_b32 %0, %1, %2"
                     :: "v"(l_wt0), "v"(g_wt0), "s"(wt_base) : "memory");
        asm volatile("global_load_async_to_lds_b32 %0, %1, %2"
                     :: "v"(l_wt1), "v"(g_wt1), "s"(wt_base) : "memory");
        // Each lane only reads back its own slots -> per-wave wait suffices.
        asm volatile("s_wait_asynccnt 0" ::: "memory");
    }

    // ---------- Per-(pixel, ref) bilinear setup, reused for all channels ----
    int   idx[Rr][4];
    float cof[Rr][4];

#pragma unroll
    for (int r = 0; r < Rr; ++r) {
        const float mvx = smem[(2 * r + 0) * 256 + tid] * 0.25f;
        const float mvy = smem[(2 * r + 1) * 256 + tid] * 0.25f;
        const float wt  = smem[(4 + r) * 256 + tid];

        const float gx = (float)w + mvx;
        const float gy = (float)h + mvy;
        const float x0f = floorf(gx);
        const float y0f = floorf(gy);
        const float fx1 = gx - x0f, fx0 = 1.0f - fx1;
        const float fy1 = gy - y0f, fy0 = 1.0f - fy1;

        const int ix0 = (int)x0f, ix1 = ix0 + 1;
        const int iy0 = (int)y0f, iy1 = iy0 + 1;

        const bool vx0 = (ix0 >= 0) & (ix0 <= Ww - 1);
        const bool vx1 = (ix1 >= 0) & (ix1 <= Ww - 1);
        const bool vy0 = (iy0 >= 0) & (iy0 <= Hh - 1);
        const bool vy1 = (iy1 >= 0) & (iy1 <= Hh - 1);

        const int cx0 = min(max(ix0, 0), Ww - 1);
        const int cx1 = min(max(ix1, 0), Ww - 1);
        const int cy0 = min(max(iy0, 0), Hh - 1);
        const int cy1 = min(max(iy1, 0), Hh - 1);

        idx[r][0] = cy0 * Ww + cx0;
        idx[r][1] = cy0 * Ww + cx1;
        idx[r][2] = cy1 * Ww + cx0;
        idx[r][3] = cy1 * Ww + cx1;

        // Fold validity and the per-pixel blend weight into the tap coeffs.
        cof[r][0] = (vy0 & vx0) ? fy0 * fx0 * wt : 0.0f;
        cof[r][1] = (vy0 & vx1) ? fy0 * fx1 * wt : 0.0f;
        cof[r][2] = (vy1 & vx0) ? fy1 * fx0 * wt : 0.0f;
        cof[r][3] = (vy1 & vx1) ? fy1 * fx1 * wt : 0.0f;
    }

    const float* __restrict__ p0 = pred + (size_t)((b * Rr + 0) * Cc) * HW;
    const float* __restrict__ p1 = pred + (size_t)((b * Rr + 1) * Cc) * HW;
    float* __restrict__ o = out + (size_t)(b * Cc) * HW + pix;

#pragma unroll 4
    for (int c = 0; c < Cc; ++c) {
        const float* __restrict__ q0 = p0 + (size_t)c * HW;
        const float* __restrict__ q1 = p1 + (size_t)c * HW;

        // Prefetch next channel's top-left tap lines (speculative TH: an OOB
        // address on the last channel is silently dropped by hardware).
        __builtin_prefetch(q0 + HW + idx[0][0], 0, 0);
        __builtin_prefetch(q1 + HW + idx[1][0], 0, 0);

        float acc;
        acc  = q0[idx[0][0]] * cof[0][0];
        acc  = fmaf(q0[idx[0][1]], cof[0][1], acc);
        acc  = fmaf(q0[idx[0][2]], cof[0][2], acc);
        acc  = fmaf(q0[idx[0][3]], cof[0][3], acc);
        acc  = fmaf(q1[idx[1][0]], cof[1][0], acc);
        acc  = fmaf(q1[idx[1][1]], cof[1][1], acc);
        acc  = fmaf(q1[idx[1][2]], cof[1][2], acc);
        acc  = fmaf(q1[idx[1][3]], cof[1][3], acc);

        o[(size_t)c * HW] = acc;   // coalesced b32 store
    }
}

extern "C" void kernel_launch(void* const* d_in, const int* in_sizes, int n_in,
                              void* d_out, int out_size, void* d_ws, size_t ws_size,
                              hipStream_t stream) {
    const float* pred = (const float*)d_in[0];  // [B,R,C,H,W] f32
    const float* mv   = (const float*)d_in[1];  // [B,R,2,H,W] f32 (quarter-pel)
    const float* wgt  = (const float*)d_in[2];  // [B,R,1,H,W] f32
    float* out = (float*)d_out;                 // [B,C,H,W]   f32

    dim3 grid(Ww / 256, Hh, Bb);                // 2 x 256 x 4 = 2048 blocks
    mc_warp_sum_kernel<<<grid, 256, 0, stream>>>(pred, mv, wgt, out);
}